// GraphConvolution_2800318677549
// MI455X (gfx1250) — compile-verified
//
#include <hip/hip_runtime.h>

typedef __attribute__((ext_vector_type(2))) float v2f;
typedef __attribute__((ext_vector_type(8))) float v8f;

#define IN_F 256
#define OUT_F 64

// ---------------------------------------------------------------------------
// Zero-fill output (harness poisons d_out with 0xAA; scatter uses atomics).
// ---------------------------------------------------------------------------
__global__ void gcn_zero_kernel(float* __restrict__ out, int n) {
    int i = blockIdx.x * blockDim.x + threadIdx.x;
    if (i < n) out[i] = 0.0f;
}

// ---------------------------------------------------------------------------
// Dense fp32 GEMM via V_WMMA_F32_16X16X4_F32.
// Each wave32 computes one 16x16 tile of support = x @ W.
//   A tile: x[row0:row0+16, k:k+4]   (16x4 f32, 2 VGPRs/lane)
//   B tile: W[k:k+4, col0:col0+16]   (4x16 f32, 2 VGPRs/lane)
// A layout (ISA 7.12.2): lanes 0-15 => M=lane, v0=K0,v1=K1; lanes 16-31 => v0=K2,v1=K3
// B layout (mirror):     lanes 0-15 => N=lane, v0=K0,v1=K1; lanes 16-31 => v0=K2,v1=K3
// C/D layout:            v[i]: lanes 0-15 => M=i, lanes 16-31 => M=i+8; N=lane&15
// ---------------------------------------------------------------------------
__global__ __launch_bounds__(256) void gcn_gemm_wmma(
        const float* __restrict__ x,
        const float* __restrict__ W,
        float* __restrict__ support,
        int n_row_tiles) {
    const int lane  = threadIdx.x & 31;
    const int wave  = (blockIdx.x << 3) + (threadIdx.x >> 5);   // 8 waves/block
    const int rowTile = wave >> 2;       // 64 cols / 16 = 4 col tiles
    const int colTile = wave & 3;
    if (rowTile >= n_row_tiles) return;  // grid sized exactly; never taken

    const int row0 = rowTile << 4;
    const int col0 = colTile << 4;
    const int m     = lane & 15;         // M (for A) / N (for B) within tile
    const int khalf = (lane >> 4) << 1;  // 0 for lanes 0-15, 2 for lanes 16-31

    v8f acc = {};

    const float* __restrict__ arow  = x + (size_t)(row0 + m) * IN_F + khalf;
    const float* __restrict__ bbase = W + (size_t)khalf * OUT_F + col0 + m;

#pragma unroll 8
    for (int k = 0; k < IN_F; k += 4) {
        // A: two consecutive K elements of this lane's row (8B aligned)
        v2f a = *(const v2f*)(arow + k);
        // B: two consecutive K rows, same output column
        v2f b;
        b.x = bbase[(size_t)k * OUT_F];
        b.y = bbase[(size_t)(k + 1) * OUT_F];
        // (neg_a, A, neg_b, B, c_mod, C, reuse_a, reuse_b)
        acc = __builtin_amdgcn_wmma_f32_16x16x4_f32(
                false, a, false, b, (short)0, acc, false, false);
    }

    // Store 16x16 f32 D tile per ISA C/D layout.
    const int mbase = row0 + ((lane >> 4) << 3);   // +8 rows for upper half-wave
    float* __restrict__ dst = support + (size_t)mbase * OUT_F + col0 + m;
#pragma unroll
    for (int i = 0; i < 8; ++i) {
        dst[(size_t)i * OUT_F] = acc[i];
    }
}

// ---------------------------------------------------------------------------
// COO SpMM scatter: one wave per edge; each lane owns a float2 feature pair.
// support is L2-resident (25.6 MB << 192 MB L2); atomics land in L2.
// ---------------------------------------------------------------------------
__global__ __launch_bounds__(256) void gcn_spmm_scatter(
        const int* __restrict__ adj_rows,
        const int* __restrict__ adj_cols,
        const float* __restrict__ adj_vals,
        const float* __restrict__ support,
        float* __restrict__ out,
        int n_edges) {
    const long long tid = (long long)blockIdx.x * blockDim.x + threadIdx.x;
    const int e = (int)(tid >> 5);
    if (e >= n_edges) return;
    const int lane = threadIdx.x & 31;

    const int   r = adj_rows[e];
    const int   c = adj_cols[e];
    const float v = adj_vals[e];

    const v2f s = *(const v2f*)(support + (size_t)c * OUT_F + (lane << 1));
    float* __restrict__ dst = out + (size_t)r * OUT_F + (lane << 1);
    // Native global_atomic_add_f32 (no CAS loop).
    unsafeAtomicAdd(dst + 0, v * s.x);
    unsafeAtomicAdd(dst + 1, v * s.y);
}

// ---------------------------------------------------------------------------
// Host-side launch. Inputs (setup_inputs order):
//   d_in[0] = x        [N_NODES * IN_F]  f32
//   d_in[1] = adj_rows [N_EDGES]         int
//   d_in[2] = adj_cols [N_EDGES]         int
//   d_in[3] = adj_vals [N_EDGES]         f32
//   d_in[4] = W        [IN_F * OUT_F]    f32
// d_ws: support [N_NODES * OUT_F] f32 (25.6 MB)
// ---------------------------------------------------------------------------
extern "C" void kernel_launch(void* const* d_in, const int* in_sizes, int n_in,
                              void* d_out, int out_size, void* d_ws, size_t ws_size,
                              hipStream_t stream) {
    const float* x        = (const float*)d_in[0];
    const int*   adj_rows = (const int*)d_in[1];
    const int*   adj_cols = (const int*)d_in[2];
    const float* adj_vals = (const float*)d_in[3];
    const float* W        = (const float*)d_in[4];
    float*       out      = (float*)d_out;
    float*       support  = (float*)d_ws;

    const int n_nodes = in_sizes[0] / IN_F;
    const int n_edges = in_sizes[1];

    // 1) zero the output accumulator
    {
        int n = out_size;
        int blocks = (n + 255) / 256;
        gcn_zero_kernel<<<blocks, 256, 0, stream>>>(out, n);
    }

    // 2) support = x @ W  (fp32 WMMA)
    {
        int row_tiles = (n_nodes + 15) / 16;        // 6250
        int waves     = row_tiles * (OUT_F / 16);   // 25000
        int blocks    = (waves + 7) / 8;            // 3125 (exact: EXEC all-1s)
        gcn_gemm_wmma<<<blocks, 256, 0, stream>>>(x, W, support, row_tiles);
    }

    // 3) out += adj @ support  (atomic scatter, one wave per edge)
    {
        long long threads = (long long)n_edges * 32;
        int blocks = (int)((threads + 255) / 256);  // 200000
        gcn_spmm_scatter<<<blocks, 256, 0, stream>>>(
            adj_rows, adj_cols, adj_vals, support, out, n_edges);
    }
}